// bahdanauDecoder_5093831213230
// MI455X (gfx1250) — compile-verified
//
#include <hip/hip_runtime.h>
#include <hip/hip_bf16.h>
#include <math.h>

// Problem dimensions (from reference)
#define VV 32000
#define HH 1024
#define BB 128
#define SS 256
#define LL 2

typedef float    v2f __attribute__((ext_vector_type(2)));
typedef float    v8f __attribute__((ext_vector_type(8)));
typedef unsigned v4u __attribute__((ext_vector_type(4)));
typedef unsigned v8u __attribute__((ext_vector_type(8)));

#define WAVES  8    // waves per block in the GEMM kernel (8 * 32 = 256 threads)
#define KCHUNK 256  // K-chunk staged into LDS by the Tensor Data Mover

// ---------------------------------------------------------------------------
// WMMA GEMM:  C[M,N] = A[M,K] @ W[N,K]^T + bias[N]   (optionally tanh)
// A row-major, W row-major [N,K] (PyTorch Linear layout), so
// C[m,n] = dot(A[m,:], W[n,:]) — both operands K-contiguous.
//
// Data movement (memory-bound design):
//  * W: streamed cold from HBM, each element read by M/16 blocks (once per 16-row M tile).
//  * A: 16xK tile per block staged into LDS by the TDM (tensor_load_to_lds),
//    double-buffered in KCHUNK slices, tracked with TENSORcnt; all 8 waves of
//    the block then feed their WMMAs from LDS (ds_load_b64) instead of 8
//    redundant global streams.
//
// f32 fragment layout (ISA 7.12.2): A 16x4: lane(m=l&15), K pair {0,1} for
// lanes 0-15, {2,3} for lanes 16-31. C/D: VGPR i holds row i+8*(l>>4),
// col = l&15.
// ---------------------------------------------------------------------------
template <int ACT> // 0 = none, 1 = tanh
__global__ __launch_bounds__(256) void wmma_gemm_f32(
    const float* __restrict__ A, const float* __restrict__ W,
    const float* __restrict__ bias, float* __restrict__ C,
    int M, int N, int K)
{
  __shared__ float tileA[2][16][KCHUNK];   // 32 KB (of 320 KB/WGP)

  const int lane = threadIdx.x & 31;
  const int wave = threadIdx.x >> 5;
  const int half = lane >> 4;     // selects K sub-pair of the f32 A/B frag
  const int l15  = lane & 15;

  const int n0 = (blockIdx.x * WAVES + wave) * 16;  // grids divide exactly
  const int m0 = blockIdx.y * 16;

  const float* wRow = W + (size_t)(n0 + l15) * K;   // B-frag: col n -> W row
  const int nch = K / KCHUNK;

  // Issue one TDM 2D-tile load: A[m0:m0+16, c*KCHUNK : (c+1)*KCHUNK] -> LDS.
  // D# per ISA ch.8: group0 = {count/lds_addr/global_addr/type=2},
  // group1 = {data_size=4B, tensor_dim0=K, tensor_dim1=16, tile_dim0=KCHUNK,
  //           tile_dim1=16, tensor_dim0_stride=K}.
  auto tdm_issue = [&](int c, int buf) {
    const unsigned lds = (unsigned)(size_t)&tileA[buf][0][0]; // LDS byte offset
    const unsigned long long ga =
        (unsigned long long)(size_t)(A + (size_t)m0 * K + (size_t)c * KCHUNK);
    v4u g0;
    g0[0] = 1u;                                   // count=1 (valid user D#)
    g0[1] = lds;                                  // lds_addr
    g0[2] = (unsigned)ga;                         // global_addr[31:0]
    g0[3] = (unsigned)(ga >> 32) | 0x80000000u;   // global_addr[56:32] | type=2
    v8u g1;
    g1[0] = 2u << 16;                             // data_size: 2 => 4 bytes
    g1[1] = ((unsigned)K & 0xffffu) << 16;        // tensor_dim0 lo
    g1[2] = ((unsigned)K >> 16) | (16u << 16);    // tensor_dim0 hi | tensor_dim1=16
    g1[3] = (unsigned)KCHUNK << 16;               // tile_dim0 = KCHUNK
    g1[4] = 16u;                                  // tile_dim1=16, tile_dim2=0
    g1[5] = (unsigned)K;                          // tensor_dim0_stride lo
    g1[6] = 0u;                                   // stride hi | dim1_stride lo
    g1[7] = 0u;
    asm volatile("tensor_load_to_lds %0, %1" :: "s"(g0), "s"(g1) : "memory");
  };

  const bool w0 = (wave == 0);
  if (w0) tdm_issue(0, 0);

  v8f acc = {};
  for (int c = 0; c < nch; ++c) {
    if (c + 1 < nch) {                  // wave-uniform control flow
      if (w0) tdm_issue(c + 1, (c + 1) & 1);
      __builtin_amdgcn_s_wait_tensorcnt(1);   // chunk c done (in-order TDM)
    } else {
      __builtin_amdgcn_s_wait_tensorcnt(0);
    }
    __syncthreads();                    // LDS tile visible to all 8 waves

    const float* aT   = &tileA[c & 1][l15][0];
    const float* wChk = wRow + (size_t)c * KCHUNK;
#pragma unroll 8
    for (int kk = 0; kk < KCHUNK; kk += 4) {
      if ((kk & 63) == 0)
        __builtin_prefetch(wChk + kk + 256, 0, 3);  // global_prefetch_b8
      const int ko = kk + 2 * half;                 // even => 8B-aligned b64
      const float2 av = *(const float2*)(aT + ko);  // ds_load_b64 from LDS
      const float2 wv = *(const float2*)(wChk + ko);
      v2f af = {av.x, av.y};
      v2f bf = {wv.x, wv.y};
      acc = __builtin_amdgcn_wmma_f32_16x16x4_f32(
          false, af, false, bf, (short)0, acc, false, false);
    }
    __syncthreads();                    // before TDM overwrites this buffer
  }

  const float bv = bias[n0 + l15];      // bias indexed by output column
#pragma unroll
  for (int i = 0; i < 8; ++i) {
    const int row = m0 + i + 8 * half;
    float v = acc[i] + bv;
    if (ACT == 1) v = tanhf(v);
    C[(size_t)row * N + (n0 + l15)] = v;
  }
}

// ---------------------------------------------------------------------------
// Embedding gather: x[b,:] = embed_table[input_ids[b], :]
// ---------------------------------------------------------------------------
__global__ void embed_kernel(const int* __restrict__ ids,
                             const float* __restrict__ table,
                             float* __restrict__ x)
{
  const int b = blockIdx.x;
  const int id = ids[b];
  for (int i = threadIdx.x; i < HH; i += blockDim.x)
    x[(size_t)b * HH + i] = table[(size_t)id * HH + i];
}

// ---------------------------------------------------------------------------
// Fused GRU gates (PyTorch order r,z,n):
// r=sig(gi0+gh0); z=sig(gi1+gh1); n=tanh(gi2 + r*gh2); h=(1-z)*n + z*h_prev
// ---------------------------------------------------------------------------
__global__ void gru_gate_kernel(const float* __restrict__ gi,
                                const float* __restrict__ gh,
                                const float* __restrict__ hprev,
                                float* __restrict__ hnew)
{
  const int idx = blockIdx.x * blockDim.x + threadIdx.x;  // over B*H
  const int b = idx / HH, j = idx % HH;
  const float* gib = gi + (size_t)b * 3 * HH;
  const float* ghb = gh + (size_t)b * 3 * HH;
  const float r = 1.f / (1.f + __expf(-(gib[j] + ghb[j])));
  const float z = 1.f / (1.f + __expf(-(gib[HH + j] + ghb[HH + j])));
  const float n = tanhf(gib[2 * HH + j] + r * ghb[2 * HH + j]);
  hnew[idx] = (1.f - z) * n + z * hprev[idx];
}

// ---------------------------------------------------------------------------
// u[e] = sum_h v[h] * W_attn[h,e]  (e in [0,2H)).  Collapses the 137-GFLOP
// attention GEMM to a vector precompute: scores = u . feats.
// ---------------------------------------------------------------------------
__global__ void attn_u_kernel(const float* __restrict__ W_attn,
                              const float* __restrict__ v,
                              float* __restrict__ u)
{
  const int e = blockIdx.x * blockDim.x + threadIdx.x;  // 2H threads
  float acc = 0.f;
  for (int h = 0; h < HH; ++h)
    acc += v[h] * W_attn[(size_t)h * (2 * HH) + e];
  u[e] = acc;
}

// ---------------------------------------------------------------------------
// scores[b,s] = dot(u2, enc[s,b,:]).  The u1.rnn_out[b] + v.b_attn part of
// the reference score is constant over s and cancels exactly in softmax
// (shift invariance), so it is omitted.  One wave per (s,b) row.
// ---------------------------------------------------------------------------
__global__ __launch_bounds__(256) void attn_scores_kernel(
    const float* __restrict__ enc, const float* __restrict__ u2,
    float* __restrict__ scores)
{
  const int lane = threadIdx.x & 31;
  const int row = blockIdx.x * 8 + (threadIdx.x >> 5);  // row = s*B + b
  const int s = row / BB, b = row % BB;
  const float4* e4 = (const float4*)(enc + (size_t)row * HH);
  const float4* u4 = (const float4*)u2;
  float acc = 0.f;
  for (int i = lane; i < HH / 4; i += 32) {
    const float4 e = e4[i], u = u4[i];
    acc += e.x * u.x + e.y * u.y + e.z * u.z + e.w * u.w;
  }
#pragma unroll
  for (int off = 16; off > 0; off >>= 1)
    acc += __shfl_down(acc, off, 32);
  if (lane == 0) scores[(size_t)b * SS + s] = acc;
}

// ---------------------------------------------------------------------------
// Row softmax over S (one block of S threads per b), written to attn output.
// ---------------------------------------------------------------------------
__global__ void softmax_kernel(const float* __restrict__ scores,
                               float* __restrict__ out)
{
  __shared__ float red[SS];
  const int b = blockIdx.x, s = threadIdx.x;
  const float x = scores[(size_t)b * SS + s];
  red[s] = x; __syncthreads();
  for (int off = SS / 2; off > 0; off >>= 1) {
    if (s < off) red[s] = fmaxf(red[s], red[s + off]);
    __syncthreads();
  }
  const float m = red[0]; __syncthreads();
  const float e = __expf(x - m);
  red[s] = e; __syncthreads();
  for (int off = SS / 2; off > 0; off >>= 1) {
    if (s < off) red[s] += red[s + off];
    __syncthreads();
  }
  out[(size_t)b * SS + s] = e / red[0];
}

// ---------------------------------------------------------------------------
// context[b,h] = sum_s w[b,s] * enc[s,b,h].  Second pass over enc hits L2
// (134 MB < 192 MB global L2).  Coalesced over h; w[b,s] scalar per block.
// ---------------------------------------------------------------------------
__global__ void context_kernel(const float* __restrict__ enc,
                               const float* __restrict__ w,
                               float* __restrict__ ctx)
{
  const int idx = blockIdx.x * blockDim.x + threadIdx.x;  // over B*H
  const int b = idx / HH, h = idx % HH;
  float acc = 0.f;
  for (int s = 0; s < SS; ++s)
    acc += w[(size_t)b * SS + s] * enc[(size_t)s * BB * HH + (size_t)b * HH + h];
  ctx[idx] = acc;
}

// concat_in[b, 0:H] = h1[b,:], concat_in[b, H:2H] = context[b,:]
__global__ void concat_build_kernel(const float* __restrict__ h1,
                                    const float* __restrict__ ctx,
                                    float* __restrict__ cin)
{
  const int idx = blockIdx.x * blockDim.x + threadIdx.x;  // over B*2H
  const int b = idx / (2 * HH), j = idx % (2 * HH);
  cin[idx] = (j < HH) ? h1[(size_t)b * HH + j] : ctx[(size_t)b * HH + (j - HH)];
}

// ---------------------------------------------------------------------------
extern "C" void kernel_launch(void* const* d_in, const int* in_sizes, int n_in,
                              void* d_out, int out_size, void* d_ws, size_t ws_size,
                              hipStream_t stream)
{
  (void)in_sizes; (void)n_in; (void)out_size; (void)ws_size;
  const int*   input_ids = (const int*)  d_in[0];
  const float* hidden    = (const float*)d_in[1];   // [L,B,H]
  const float* enc       = (const float*)d_in[2];   // [S,B,H]
  const float* embed     = (const float*)d_in[3];   // [V,H]
  const float* W_attn    = (const float*)d_in[4];   // [H,2H]
  /* b_attn d_in[5] unused: cancels in softmax (shift invariance) */
  const float* v_vec     = (const float*)d_in[6];   // [H]
  const float* w_ih0     = (const float*)d_in[7];
  const float* w_hh0     = (const float*)d_in[8];
  const float* b_ih0     = (const float*)d_in[9];
  const float* b_hh0     = (const float*)d_in[10];
  const float* w_ih1     = (const float*)d_in[11];
  const float* w_hh1     = (const float*)d_in[12];
  const float* b_ih1     = (const float*)d_in[13];
  const float* b_hh1     = (const float*)d_in[14];
  const float* W_concat  = (const float*)d_in[15];  // [H,2H]
  const float* b_concat  = (const float*)d_in[16];
  const float* W_out     = (const float*)d_in[17];  // [V,H]
  const float* b_out     = (const float*)d_in[18];

  float* out       = (float*)d_out;
  float* out_logit = out;                                   // [B,V]
  float* out_h0    = out + (size_t)BB * VV;                 // new_hidden[0]
  float* out_h1    = out_h0 + (size_t)BB * HH;              // new_hidden[1]
  float* out_attn  = out + (size_t)BB * VV + (size_t)LL * BB * HH;  // [B,1,S]

  float* ws     = (float*)d_ws;                 // ~5.9 MB total
  float* x      = ws;                           // B*H
  float* gi     = x + (size_t)BB * HH;          // B*3H
  float* gh     = gi + (size_t)3 * BB * HH;     // B*3H
  float* u      = gh + (size_t)3 * BB * HH;     // 2H
  float* scores = u + 2 * HH;                   // B*S
  float* ctx    = scores + (size_t)BB * SS;     // B*H
  float* cin    = ctx + (size_t)BB * HH;        // B*2H
  float* cout_  = cin + (size_t)2 * BB * HH;    // B*H

#define LAUNCH_GEMM(ACT, A_, W_, BIAS_, C_, M_, N_, K_)                         \
  wmma_gemm_f32<ACT><<<dim3((N_) / (16 * WAVES), (M_) / 16), 256, 0, stream>>>( \
      A_, W_, BIAS_, C_, M_, N_, K_)

  // 1) Embedding lookup
  embed_kernel<<<BB, 256, 0, stream>>>(input_ids, embed, x);

  // 2) GRU layer 0 (h0 written straight into new_hidden slot of d_out)
  LAUNCH_GEMM(0, x, w_ih0, b_ih0, gi, BB, 3 * HH, HH);
  LAUNCH_GEMM(0, hidden, w_hh0, b_hh0, gh, BB, 3 * HH, HH);
  gru_gate_kernel<<<(BB * HH) / 256, 256, 0, stream>>>(gi, gh, hidden, out_h0);

  // 3) GRU layer 1
  LAUNCH_GEMM(0, out_h0, w_ih1, b_ih1, gi, BB, 3 * HH, HH);
  LAUNCH_GEMM(0, hidden + (size_t)BB * HH, w_hh1, b_hh1, gh, BB, 3 * HH, HH);
  gru_gate_kernel<<<(BB * HH) / 256, 256, 0, stream>>>(
      gi, gh, hidden + (size_t)BB * HH, out_h1);

  // 4) Attention: u = W_attn^T v, scores, softmax, context
  attn_u_kernel<<<(2 * HH) / 256, 256, 0, stream>>>(W_attn, v_vec, u);
  attn_scores_kernel<<<(SS * BB) / 8, 256, 0, stream>>>(enc, u + HH, scores);
  softmax_kernel<<<BB, SS, 0, stream>>>(scores, out_attn);
  context_kernel<<<(BB * HH) / 256, 256, 0, stream>>>(enc, out_attn, ctx);

  // 5) concat -> tanh(Linear) -> output logits
  concat_build_kernel<<<(2 * BB * HH) / 256, 256, 0, stream>>>(out_h1, ctx, cin);
  LAUNCH_GEMM(1, cin, W_concat, b_concat, cout_, BB, HH, 2 * HH);
  LAUNCH_GEMM(0, cout_, W_out, b_out, out_logit, BB, VV, HH);

#undef LAUNCH_GEMM
}